// GraphConvolution_55224689492023
// MI455X (gfx1250) — compile-verified
//
#include <hip/hip_runtime.h>
#include <hip/hip_bf16.h>

// ---------------------------------------------------------------------------
// GraphConvolution on MI455X (gfx1250), f32 end to end.
//   s = sqrt(colsum(adj))
//   G = s .* (feature @ weight)                       [8192 x 128]
//   out = s .* (adj@G + adjT@G + G) + bias            [8192 x 128]
// Dominant cost: 3 streaming passes over adj (268MB each) at 23.3 TB/s.
// G (4MB) stays hot in the 192MB L2. Matrix math via V_WMMA_F32_16X16X4_F32.
// ---------------------------------------------------------------------------

typedef float v2f __attribute__((ext_vector_type(2)));
typedef float v8f __attribute__((ext_vector_type(8)));

#define GC_N    8192
#define GC_INF  256
#define GC_OUTF 128

// ---------------------------------------------------------------- K0: zero s
__global__ void gc_zero_s(float* __restrict__ sraw) {
    int j = blockIdx.x * 256 + threadIdx.x;
    if (j < GC_N) sraw[j] = 0.0f;
}

// ------------------------------------------------ K1: column sums (atomic fp)
// grid (32, 16): 256 cols per block.x, 512 rows per block.y  -> 512 blocks
__global__ __launch_bounds__(256)
void gc_colsum(const float* __restrict__ adj, float* __restrict__ sraw) {
    const int j  = blockIdx.x * 256 + threadIdx.x;
    const int r0 = blockIdx.y * 512;
    float sum = 0.0f;
    #pragma unroll 8
    for (int i = 0; i < 512; ++i)
        sum += adj[(size_t)(r0 + i) * GC_N + j];
    __hip_atomic_fetch_add(&sraw[j], sum, __ATOMIC_RELAXED,
                           __HIP_MEMORY_SCOPE_AGENT);
}

// ---------------------------------------------------------------- K2: sqrt s
__global__ void gc_sqrt_s(float* __restrict__ s) {
    int j = blockIdx.x * 256 + threadIdx.x;
    if (j < GC_N) s[j] = sqrtf(s[j]);
}

// ------------------------------------- K3: G = s .* (feature @ weight)
// one block (128 threads) per row; weight (128KB) is cache-hot.
__global__ __launch_bounds__(128)
void gc_featmul(const float* __restrict__ feature,
                const float* __restrict__ weight,
                const float* __restrict__ s,
                float* __restrict__ G) {
    __shared__ float fs[GC_INF];
    const int i = blockIdx.x;
    const int c = threadIdx.x;
    fs[c]       = feature[(size_t)i * GC_INF + c];
    fs[c + 128] = feature[(size_t)i * GC_INF + c + 128];
    __syncthreads();
    float acc = 0.0f;
    #pragma unroll 8
    for (int f = 0; f < GC_INF; ++f)
        acc = fmaf(fs[f], weight[f * GC_OUTF + c], acc);
    G[(size_t)i * GC_OUTF + c] = s[i] * acc;
}

// ------------------------------------- K4: P = adj @ G  /  Q = adjT @ G
// grid (128, 2): block.x picks the 64-row strip, block.y picks orientation.
// Block = 256 threads = 8 waves; wave (w%4 -> 16-row group, w/4 -> 64-col half)
// Tiles staged through LDS; fragments follow the CDNA5 32-bit WMMA lane maps:
//   A 16x4 : lane L<16 -> row L, K={0,1};  lane 16+L -> row L, K={2,3}
//   B 4x16 : lane L<16 -> col L, K={0,1};  lane 16+L -> col L, K={2,3}
//   C 16x16: vgpr e, lanes 0-15 -> row e, lanes 16-31 -> row e+8, col = L%16
__global__ __launch_bounds__(256)
void gc_gemm_adj(const float* __restrict__ adj,
                 const float* __restrict__ G,
                 float* __restrict__ P,
                 float* __restrict__ Q) {
    constexpr int KC   = 32;
    constexpr int MBLK = 64;
    constexpr int LDA  = KC + 1;          // pad to 33 floats: kills bank conflicts

    __shared__ float As[MBLK * LDA];      // [r][kk]  ~8.4 KB
    __shared__ float Bs[KC * GC_OUTF];    // [kk][n]  16 KB

    const int tid  = threadIdx.x;
    const int wave = tid >> 5;
    const int lane = tid & 31;
    const int lh   = lane >> 4;           // K-half select
    const int lm   = lane & 15;           // row/col within 16

    const int  R0    = blockIdx.x * MBLK;
    const bool trans = (blockIdx.y != 0);
    float* __restrict__ out = trans ? Q : P;

    const int wr = (wave & 3) * 16;       // wave row base within block
    const int wc = (wave >> 2) * 64;      // wave col base within 128

    v8f acc[4] = {};

    for (int k = 0; k < GC_N; k += KC) {
        __syncthreads();
        if (!trans) {
            // As[r][kk] = adj[R0+r][k+kk]; 64 rows x 8 float4 chunks = 512
            #pragma unroll
            for (int it = 0; it < 2; ++it) {
                int idx = tid + it * 256;
                int r   = idx >> 3;
                int kk  = (idx & 7) << 2;
                float4 v = *reinterpret_cast<const float4*>(
                    adj + (size_t)(R0 + r) * GC_N + k + kk);
                As[r * LDA + kk + 0] = v.x;
                As[r * LDA + kk + 1] = v.y;
                As[r * LDA + kk + 2] = v.z;
                As[r * LDA + kk + 3] = v.w;
            }
        } else {
            // As[r][kk] = adj[k+kk][R0+r]; 32 kk-rows x 16 float4 chunks = 512
            #pragma unroll
            for (int it = 0; it < 2; ++it) {
                int idx = tid + it * 256;
                int kk  = idx >> 4;
                int r   = (idx & 15) << 2;
                float4 v = *reinterpret_cast<const float4*>(
                    adj + (size_t)(k + kk) * GC_N + R0 + r);
                As[(r + 0) * LDA + kk] = v.x;
                As[(r + 1) * LDA + kk] = v.y;
                As[(r + 2) * LDA + kk] = v.z;
                As[(r + 3) * LDA + kk] = v.w;
            }
        }
        // Bs[kk][n] = G[k+kk][n]; 4096 floats = 1024 float4 (L2-hot source)
        #pragma unroll
        for (int it = 0; it < 4; ++it) {
            int idx = tid + it * 256;
            float4 v = *reinterpret_cast<const float4*>(
                G + (size_t)k * GC_OUTF + idx * 4);
            *reinterpret_cast<float4*>(Bs + idx * 4) = v;
        }
        __syncthreads();

        #pragma unroll
        for (int kk0 = 0; kk0 < KC; kk0 += 4) {
            v2f a;
            const float* ap = As + (wr + lm) * LDA + kk0 + lh * 2;
            a.x = ap[0];
            a.y = ap[1];
            #pragma unroll
            for (int t = 0; t < 4; ++t) {
                v2f b;
                const float* bp = Bs + (kk0 + lh * 2) * GC_OUTF + wc + t * 16 + lm;
                b.x = bp[0];
                b.y = bp[GC_OUTF];
                acc[t] = __builtin_amdgcn_wmma_f32_16x16x4_f32(
                    false, a, false, b, (short)0, acc[t], false, false);
            }
        }
    }

    // write back C tiles (row-major [row][col])
    #pragma unroll
    for (int t = 0; t < 4; ++t) {
        #pragma unroll
        for (int e = 0; e < 8; ++e) {
            int row = R0 + wr + lh * 8 + e;
            int col = wc + t * 16 + lm;
            out[(size_t)row * GC_OUTF + col] = acc[t][e];
        }
    }
}

// ------------------------------------- K5: out = s.*(P+Q+G) + bias
__global__ __launch_bounds__(256)
void gc_epilogue(const float* __restrict__ P, const float* __restrict__ Q,
                 const float* __restrict__ G, const float* __restrict__ s,
                 const float* __restrict__ bias, float* __restrict__ out) {
    size_t idx = (size_t)blockIdx.x * 256 + threadIdx.x;
    int i = (int)(idx >> 7);
    int c = (int)(idx & 127);
    out[idx] = s[i] * (P[idx] + Q[idx] + G[idx]) + bias[c];
}

// ---------------------------------------------------------------------------
extern "C" void kernel_launch(void* const* d_in, const int* in_sizes, int n_in,
                              void* d_out, int out_size, void* d_ws, size_t ws_size,
                              hipStream_t stream) {
    const float* adj     = (const float*)d_in[0];
    const float* feature = (const float*)d_in[1];
    const float* weight  = (const float*)d_in[2];
    const float* bias    = (const float*)d_in[3];
    float*       out     = (float*)d_out;

    // workspace carve-out (floats): s | G | P | Q  => ~12.6 MB
    float* W = (float*)d_ws;
    float* s = W;
    float* G = W + GC_N;
    float* P = G + (size_t)GC_N * GC_OUTF;
    float* Q = P + (size_t)GC_N * GC_OUTF;

    gc_zero_s  <<<GC_N / 256, 256, 0, stream>>>(s);
    gc_colsum  <<<dim3(GC_N / 256, 16), 256, 0, stream>>>(adj, s);
    gc_sqrt_s  <<<GC_N / 256, 256, 0, stream>>>(s);
    gc_featmul <<<GC_N, 128, 0, stream>>>(feature, weight, s, G);
    gc_gemm_adj<<<dim3(GC_N / 64, 2), 256, 0, stream>>>(adj, G, P, Q);
    gc_epilogue<<<(GC_N * GC_OUTF) / 256, 256, 0, stream>>>(P, Q, G, s, bias, out);
}